// _RMSNormInt_1254130450889
// MI455X (gfx1250) — compile-verified
//
#include <hip/hip_runtime.h>
#include <stdint.h>

// Q16.16 fixed-point RMSNorm, rows of 4096 fp32, staged through LDS with the
// CDNA5 Tensor Data Mover. Double-buffered TDM pipeline: 8 rows per block,
// load row i+1 / store row i-1 overlap compute of row i.
//
// Roofline: 256 MiB in + 256 MiB out @ 23.3 TB/s ~= 22 us -> memory bound.
// Row-uniform divisor lets us replace i64 division with a double reciprocal
// plus exact floor fixup, keeping VALU work far below the HBM ceiling.

#define D          4096
#define BLOCK      512
#define PER_THREAD (D / BLOCK)   // 8
#define NWAVES     (BLOCK / 32)  // 16
#define RPB        8             // rows per block (pipelined)

typedef uint32_t u32x4 __attribute__((ext_vector_type(4)));
typedef int32_t  i32x4 __attribute__((ext_vector_type(4)));
typedef int32_t  i32x8 __attribute__((ext_vector_type(8)));

#if __has_builtin(__builtin_amdgcn_tensor_load_to_lds) && \
    __has_builtin(__builtin_amdgcn_tensor_store_from_lds) && \
    __has_builtin(__builtin_amdgcn_s_wait_tensorcnt)
#define HAVE_TDM 1
#else
#define HAVE_TDM 0
#endif

#if HAVE_TDM
// D# group0: count=1 | lds_addr | 57-bit global addr [120:64] | type=2.
__device__ __forceinline__ u32x4 tdm_group0(uint32_t lds_addr, uint64_t gaddr) {
    u32x4 g0;
    g0[0] = 1u;
    g0[1] = lds_addr;
    g0[2] = (uint32_t)gaddr;
    g0[3] = ((uint32_t)(gaddr >> 32) & 0x01FFFFFFu) | (2u << 30);
    return g0;
}

// D# group1: data_size=4B, tensor_dim0 = tile_dim0 = D, tensor/tile_dim1 = 1,
// dim0_stride = D. No multicast / padding / iteration / atomic-barrier.
__device__ __forceinline__ i32x8 tdm_group1() {
    i32x8 g1;
    g1[0] = (int)(2u << 16);          // data_size = 2 -> 4 bytes
    g1[1] = (int)((uint32_t)D << 16); // tensor_dim0[15:0] @ dw1[31:16]
    g1[2] = (int)(1u << 16);          // tensor_dim1 = 1
    g1[3] = (int)((uint32_t)D << 16); // tile_dim0 = D @ dw3[31:16]
    g1[4] = 1;                        // tile_dim1 = 1, tile_dim2 = 0
    g1[5] = D;                        // tensor_dim0_stride lo32
    g1[6] = 0;
    g1[7] = 0;
    return g1;
}

__device__ __forceinline__ void tdm_load_row(uint32_t lds_addr, uint64_t gaddr) {
    u32x4 g0 = tdm_group0(lds_addr, gaddr);
    i32x8 g1 = tdm_group1();
    i32x4 z4 = {0, 0, 0, 0};
#if __clang_major__ >= 23
    i32x8 z8 = {0, 0, 0, 0, 0, 0, 0, 0};
    __builtin_amdgcn_tensor_load_to_lds(g0, g1, z4, z4, z8, 0);
#else
    __builtin_amdgcn_tensor_load_to_lds(g0, g1, z4, z4, 0);
#endif
}

__device__ __forceinline__ void tdm_store_row(uint32_t lds_addr, uint64_t gaddr) {
    u32x4 g0 = tdm_group0(lds_addr, gaddr);
    i32x8 g1 = tdm_group1();
    i32x4 z4 = {0, 0, 0, 0};
#if __clang_major__ >= 23
    i32x8 z8 = {0, 0, 0, 0, 0, 0, 0, 0};
    __builtin_amdgcn_tensor_store_from_lds(g0, g1, z4, z4, z8, 0);
#else
    __builtin_amdgcn_tensor_store_from_lds(g0, g1, z4, z4, 0);
#endif
}
#endif // HAVE_TDM

__global__ __launch_bounds__(BLOCK)
void rmsnorm_q16_kernel(const float* __restrict__ x,
                        const float* __restrict__ gamma,
                        float* __restrict__ out,
                        int nrows) {
    __shared__ float     tile[2][D];    // 2 x 16 KB double-buffered row tiles
    __shared__ int       gfix[D];       // 16 KB quantized gamma (per block)
    __shared__ long long wsum[NWAVES];
    __shared__ long long total;

    const int tid  = threadIdx.x;
    const int base = blockIdx.x * RPB;
    const int n    = min(RPB, nrows - base);
    if (n <= 0) return;

    // Quantize gamma once per block (L2-resident; amortized over RPB rows).
    for (int k = tid; k < D; k += BLOCK)
        gfix[k] = (int)rintf(gamma[k] * 65536.0f);

#if HAVE_TDM
    if (tid == 0) {                     // prologue: prefetch rows 0 and 1
        tdm_load_row((uint32_t)(uintptr_t)(void*)&tile[0][0],
                     (uint64_t)(uintptr_t)(x + (size_t)base * D));
        if (n > 1)
            tdm_load_row((uint32_t)(uintptr_t)(void*)&tile[1][0],
                         (uint64_t)(uintptr_t)(x + (size_t)(base + 1) * D));
    }
#endif

    for (int i = 0; i < n; ++i) {
        float* buf = &tile[i & 1][0];
        const size_t row = (size_t)(base + i) * D;

#if HAVE_TDM
        if (tid == 0) {
            // Same-wave tensor ops complete in order. Ops issued after L(i):
            // S(i-1) [i>0] and L(i+1) [i+1<n]. Waiting tensorcnt <= that count
            // guarantees L(i) (and everything older) is complete.
            const bool more = (i + 1 < n);
            if (i == 0) { if (more) __builtin_amdgcn_s_wait_tensorcnt(1);
                          else      __builtin_amdgcn_s_wait_tensorcnt(0); }
            else        { if (more) __builtin_amdgcn_s_wait_tensorcnt(2);
                          else      __builtin_amdgcn_s_wait_tensorcnt(1); }
        }
        __syncthreads();                // B1: tile i ready (also covers gfix)
#else
        for (int k = tid; k < D; k += BLOCK) buf[k] = x[row + k];
        __syncthreads();
#endif

        // ---- pass 1: quantize, exact int64 sum of floor(x^2 / 2^16)
        int       xi[PER_THREAD];
        long long ssum = 0;
#pragma unroll
        for (int k = 0; k < PER_THREAD; ++k) {
            const int idx = k * BLOCK + tid;
            const int v   = (int)rintf(buf[idx] * 65536.0f);  // RNE == jnp.round
            xi[k] = v;
            ssum += (((long long)v * (long long)v) >> 16);
        }
#pragma unroll
        for (int off = 16; off > 0; off >>= 1)
            ssum += __shfl_down(ssum, off, 32);
        if ((tid & 31) == 0) wsum[tid >> 5] = ssum;
        __syncthreads();                // B2
        if (tid == 0) {
            long long t = 0;
#pragma unroll
            for (int w = 0; w < NWAVES; ++w) t += wsum[w];
            total = t;
        }
        __syncthreads();                // B3

        const long long mean_sq = total >> 12;  // floor(/4096)
        const long long rms =
            (long long)floor(sqrt((double)(mean_sq << 16))) + 1;
        const double inv = 1.0 / (double)rms;   // row-uniform reciprocal

        // ---- pass 2: y = floor(x_fix * gamma_fix / rms) via reciprocal +
        // exact fixup (|num| <= 2^37 so (double)num is exact; q0 within +-1).
#pragma unroll
        for (int k = 0; k < PER_THREAD; ++k) {
            const int idx = k * BLOCK + tid;
            const long long num = (long long)xi[k] * (long long)gfix[idx];
            long long q = (long long)floor((double)num * inv);
            long long r = num - q * rms;
            if (r < 0)    { q -= 1; r += rms; }
            if (r < 0)    { q -= 1; r += rms; }
            if (r >= rms) { q += 1; r -= rms; }
            if (r >= rms) { q += 1; }
            const float y = (float)q * (1.0f / 65536.0f);
#if HAVE_TDM
            buf[idx] = y;               // stage result in the same tile
#else
            out[row + idx] = y;
#endif
        }

#if HAVE_TDM
        __syncthreads();                // B4: tile i fully written
        if (tid == 0) {
            tdm_store_row((uint32_t)(uintptr_t)(void*)buf,
                          (uint64_t)(uintptr_t)(out + row));
            if (i + 2 < n)              // reload same tile behind its store;
                tdm_load_row((uint32_t)(uintptr_t)(void*)buf,   // same-wave
                             (uint64_t)(uintptr_t)(x + row + 2 * D)); // order
        }
#endif
    }
    // Final tensor store drains via S_ENDPGM's implicit wait-idle.
}

extern "C" void kernel_launch(void* const* d_in, const int* in_sizes, int n_in,
                              void* d_out, int out_size, void* d_ws, size_t ws_size,
                              hipStream_t stream) {
    (void)n_in; (void)out_size; (void)d_ws; (void)ws_size;
    const float* x     = (const float*)d_in[0];   // (4, 4096, 4096) fp32
    const float* gamma = (const float*)d_in[1];   // (4096,) fp32
    float*       out   = (float*)d_out;

    const int rows   = in_sizes[0] / D;           // 16384
    const int blocks = (rows + RPB - 1) / RPB;    // 2048
    rmsnorm_q16_kernel<<<blocks, BLOCK, 0, stream>>>(x, gamma, out, rows);
}